// TransConv_68865505624456
// MI455X (gfx1250) — compile-verified
//
#include <hip/hip_runtime.h>

#define HEADS 4
#define ATT 128
#define IN_FEAT 256

typedef float v2f __attribute__((ext_vector_type(2)));
typedef float v8f __attribute__((ext_vector_type(8)));

// ---------------------------------------------------------------------------
// float atomic max via int/uint ordering trick (emits global_atomic_max_i32 /
// global_atomic_min_u32 -- no CAS loop)
// ---------------------------------------------------------------------------
__device__ __forceinline__ void atomicMaxFloat(float* addr, float val) {
  if (val >= 0.0f) {
    atomicMax((int*)addr, __float_as_int(val));
  } else {
    atomicMin((unsigned int*)addr, __float_as_uint(val));
  }
}

// ---------------------------------------------------------------------------
// Init: zero the output accumulator, set segment-max to -inf, denom to 0
// ---------------------------------------------------------------------------
__global__ void tc_init_kernel(float* __restrict__ out, float* __restrict__ mx,
                               float* __restrict__ denom, int N) {
  int idx = blockIdx.x * blockDim.x + threadIdx.x;
  if (idx < N * ATT) out[idx] = 0.0f;
  if (idx < N * HEADS) {
    mx[idx] = -__builtin_inff();
    denom[idx] = 0.0f;
  }
}

// ---------------------------------------------------------------------------
// QKV projection: out[n, o] = sum_k x[n,k] * W[o,k] + bias[o]
// One wave32 computes a 16-row x 128-col tile via V_WMMA_F32_16X16X4_F32.
// W K-slices are staged through LDS (stride 17 to avoid bank conflicts).
//
// A (16x4 f32) lane layout: lane l -> row l%16, K pair 2*(l/16)+{0,1}
// B (4x16 f32) lane layout: lane l -> W row (tile*16 + l%16), same K pair
// C/D (16x16 f32): VGPR r, lane l -> (M = r + 8*(l/16), Ncol = l%16)
// ---------------------------------------------------------------------------
__global__ __launch_bounds__(256) void tc_qkv_gemm_kernel(
    const float* __restrict__ x, const float* __restrict__ W,
    const float* __restrict__ bias, float* __restrict__ out, int N) {
  __shared__ float Wl[128 * 17];

  const int lane = threadIdx.x & 31;
  const int wave = threadIdx.x >> 5;
  const int hi = lane >> 4;   // which K pair of the 16x16x4 step
  const int lm = lane & 15;   // row/col within tile

  const int row0 = (blockIdx.x * 8 + wave) * 16;
  int arow = row0 + lm;
  if (arow > N - 1) arow = N - 1;  // clamp; stores are guarded below
  const float* xrow = x + (size_t)arow * IN_FEAT;

  v8f acc[8];
#pragma unroll
  for (int t = 0; t < 8; ++t)
#pragma unroll
    for (int r = 0; r < 8; ++r) acc[t][r] = 0.0f;

  for (int k0 = 0; k0 < IN_FEAT; k0 += 16) {
    // cooperative stage of W[:, k0:k0+16] into LDS (2048 floats, 8/thread)
    {
      const int o = threadIdx.x >> 1;
      const int kk = (threadIdx.x & 1) * 8;
      const float* g = W + (size_t)o * IN_FEAT + k0 + kk;
      float* s = &Wl[o * 17 + kk];
#pragma unroll
      for (int j = 0; j < 8; ++j) s[j] = g[j];
    }
    __syncthreads();

#pragma unroll
    for (int kk = 0; kk < 16; kk += 4) {
      v2f a = *(const v2f*)(xrow + k0 + kk + hi * 2);
#pragma unroll
      for (int t = 0; t < 8; ++t) {
        const int widx = (t * 16 + lm) * 17 + kk + hi * 2;
        v2f b;
        b.x = Wl[widx];
        b.y = Wl[widx + 1];
        acc[t] = __builtin_amdgcn_wmma_f32_16x16x4_f32(
            false, a, false, b, (short)0, acc[t], false, false);
      }
    }
    __syncthreads();
  }

  if (row0 < N) {
#pragma unroll
    for (int t = 0; t < 8; ++t) {
      const int col = t * 16 + lm;
      const float bb = bias[col];
#pragma unroll
      for (int r = 0; r < 8; ++r) {
        const int row = row0 + r + hi * 8;
        if (row < N) out[(size_t)row * ATT + col] = acc[t][r] + bb;
      }
    }
  }
}

// ---------------------------------------------------------------------------
// Pass A: per-edge scores (dot over d_k=32 per head) + segment max by src.
// One wave32 per edge; lane l handles 4 contiguous features (head = l/8).
// ---------------------------------------------------------------------------
__global__ __launch_bounds__(256) void tc_edge_score_kernel(
    const int* __restrict__ src_idx, const int* __restrict__ dst_idx,
    const float* __restrict__ q, const float* __restrict__ k,
    float* __restrict__ scores, float* __restrict__ mx, int E) {
  const int lane = threadIdx.x & 31;
  const int waveId = (blockIdx.x * blockDim.x + threadIdx.x) >> 5;
  const int nWaves = (gridDim.x * blockDim.x) >> 5;
  const float inv_sqrt_dk = 0.17677669529663687f;  // 1/sqrt(32)

  for (int e = waveId; e < E; e += nWaves) {
    const int s = src_idx[e];
    const int d = dst_idx[e];
    const float4 qv = *(const float4*)(q + (size_t)s * ATT + lane * 4);
    const float4 kv = *(const float4*)(k + (size_t)d * ATT + lane * 4);
    float p = qv.x * kv.x + qv.y * kv.y + qv.z * kv.z + qv.w * kv.w;
    // butterfly reduce within each 8-lane head group
    p += __shfl_xor(p, 1, 32);
    p += __shfl_xor(p, 2, 32);
    p += __shfl_xor(p, 4, 32);
    const float sc = p * inv_sqrt_dk;
    const int h = lane >> 3;
    if ((lane & 7) == 0) {
      scores[(size_t)e * HEADS + h] = sc;
      atomicMaxFloat(&mx[(size_t)s * HEADS + h], sc);
    }
  }
}

// ---------------------------------------------------------------------------
// Pass B: ex = exp(score - max); denom[src,h] += ex;
//         out[src, h, :] += ex * v[dst, h, :]   (unnormalized numerator)
// ---------------------------------------------------------------------------
__global__ __launch_bounds__(256) void tc_edge_accum_kernel(
    const int* __restrict__ src_idx, const int* __restrict__ dst_idx,
    const float* __restrict__ v, const float* __restrict__ scores,
    const float* __restrict__ mx, float* __restrict__ denom,
    float* __restrict__ out, int E) {
  const int lane = threadIdx.x & 31;
  const int waveId = (blockIdx.x * blockDim.x + threadIdx.x) >> 5;
  const int nWaves = (gridDim.x * blockDim.x) >> 5;

  for (int e = waveId; e < E; e += nWaves) {
    const int s = src_idx[e];
    const int d = dst_idx[e];
    const int h = lane >> 3;
    const float sc = scores[(size_t)e * HEADS + h];
    const float m = mx[(size_t)s * HEADS + h];
    const float ex = __expf(sc - m);
    if ((lane & 7) == 0) {
      unsafeAtomicAdd(&denom[(size_t)s * HEADS + h], ex);
    }
    const float4 vv = *(const float4*)(v + (size_t)d * ATT + lane * 4);
    float* ob = out + (size_t)s * ATT + lane * 4;
    unsafeAtomicAdd(ob + 0, ex * vv.x);
    unsafeAtomicAdd(ob + 1, ex * vv.y);
    unsafeAtomicAdd(ob + 2, ex * vv.z);
    unsafeAtomicAdd(ob + 3, ex * vv.w);
  }
}

// ---------------------------------------------------------------------------
// Pass C: normalize. Empty segments (denom==0) -> 0, matching segment_sum.
// ---------------------------------------------------------------------------
__global__ void tc_finalize_kernel(float* __restrict__ out,
                                   const float* __restrict__ denom, int N) {
  int idx = blockIdx.x * blockDim.x + threadIdx.x;
  if (idx >= N * ATT) return;
  const int n = idx >> 7;          // /128
  const int h = (idx >> 5) & 3;    // (%128)/32
  const float dn = denom[n * HEADS + h];
  const float val = out[idx];
  out[idx] = (dn > 0.0f) ? (val / dn) : 0.0f;
}

// ---------------------------------------------------------------------------
extern "C" void kernel_launch(void* const* d_in, const int* in_sizes, int n_in,
                              void* d_out, int out_size, void* d_ws,
                              size_t ws_size, hipStream_t stream) {
  const float* x  = (const float*)d_in[0];
  const int* edge = (const int*)d_in[1];
  const float* Qw = (const float*)d_in[2];
  const float* Qb = (const float*)d_in[3];
  const float* Kw = (const float*)d_in[4];
  const float* Kb = (const float*)d_in[5];
  const float* Vw = (const float*)d_in[6];
  const float* Vb = (const float*)d_in[7];

  const int N = in_sizes[0] / IN_FEAT;  // 50000
  const int E = in_sizes[1] / 2;        // 1.6M
  const int* srcI = edge;
  const int* dstI = edge + E;

  float* ws = (float*)d_ws;
  float* q      = ws;
  float* kbuf   = q + (size_t)N * ATT;
  float* vbuf   = kbuf + (size_t)N * ATT;
  float* scores = vbuf + (size_t)N * ATT;
  float* mx     = scores + (size_t)E * HEADS;
  float* denom  = mx + (size_t)N * HEADS;
  float* out    = (float*)d_out;

  // init out accumulator / segment max / denom
  {
    int total = N * ATT;
    tc_init_kernel<<<(total + 255) / 256, 256, 0, stream>>>(out, mx, denom, N);
  }

  // Q/K/V projections (WMMA f32 16x16x4)
  {
    const int tiles = (N + 15) / 16;
    dim3 grid((tiles + 7) / 8);
    tc_qkv_gemm_kernel<<<grid, 256, 0, stream>>>(x, Qw, Qb, q, N);
    tc_qkv_gemm_kernel<<<grid, 256, 0, stream>>>(x, Kw, Kb, kbuf, N);
    tc_qkv_gemm_kernel<<<grid, 256, 0, stream>>>(x, Vw, Vb, vbuf, N);
  }

  // edge passes: one wave32 per edge, grid-stride
  tc_edge_score_kernel<<<8192, 256, 0, stream>>>(srcI, dstI, q, kbuf, scores,
                                                 mx, E);
  tc_edge_accum_kernel<<<8192, 256, 0, stream>>>(srcI, dstI, vbuf, scores, mx,
                                                 denom, out, E);

  // normalize
  tc_finalize_kernel<<<(N * ATT + 255) / 256, 256, 0, stream>>>(out, denom, N);
}